// GAT_55860344651795
// MI455X (gfx1250) — compile-verified
//
#include <hip/hip_runtime.h>
#include <hip/hip_bf16.h>
#include <math.h>

// Problem constants (match reference)
#define NN   10000
#define EE   100000
#define FIN  128
#define CCH  120            // channels per head
#define HH   16             // heads (layers 1-2)
#define HC   (HH * CCH)     // 1920
#define KT   32             // K-tile staged in LDS per block iteration
#define BSTRIDE 34          // LDS row stride (even -> 8B-aligned b64 reads, no conflicts)

typedef float v2f __attribute__((ext_vector_type(2)));
typedef float v8f __attribute__((ext_vector_type(8)));

static __device__ __forceinline__ v8f wmma4(v2f a, v2f b, v8f c) {
    // D = A(16x4,f32) * B(4x16,f32) + C(16x16,f32)
    return __builtin_amdgcn_wmma_f32_16x16x4_f32(
        /*neg_a=*/false, a, /*neg_b=*/false, b,
        /*c_mod=*/(short)0, c, /*reuse_a=*/false, /*reuse_b=*/false);
}

// ---------------------------------------------------------------------------
// WMMA GEMM: D[M,Nc] = A[M,K] @ B[K,Nc]  (+bias, +relu optional epilogue)
// Block: 256 threads = 8 waves; wave w owns M-tile (blockIdx.y*8+w), all waves
// share the same 64 output columns. B K-tile (KT x 64) is staged in LDS once
// per block (8x reuse), transposed so each lane reads its {k,k+1} pair as one
// aligned b64. K loop is split into an unguarded main part (pointer-advance,
// immediate offsets) and a single clamped tail tile -> branchless everywhere.
// Requires K % 4 == 0, M % 16 == 0.
// ---------------------------------------------------------------------------
template <bool BIAS, bool RELU>
__global__ void __launch_bounds__(256)
gat_gemm_wmma(const float* __restrict__ A,
              const float* __restrict__ Bm,
              const float* __restrict__ bias,
              float* __restrict__ D,
              int M, int Nc, int K) {
    __shared__ float bsh[64 * BSTRIDE];

    const int lane  = threadIdx.x & 31;
    const int wave  = threadIdx.x >> 5;
    const int half2 = (lane >> 4) << 1;   // 0 or 2: which K pair this half-wave owns
    const int l15   = lane & 15;
    const int n0    = blockIdx.x * 64;
    const int numTileM = M >> 4;
    int tileM = blockIdx.y * 8 + wave;
    const bool validM = (tileM < numTileM);
    if (!validM) tileM = numTileM - 1;     // clamp: keep wave alive for barriers
    const float* Arow = A + (size_t)(tileM * 16 + l15) * K;

    // Staging-loop lane mapping (consecutive threads -> consecutive cols)
    const int sc = threadIdx.x & 63;                 // local col 0..63
    const int sk = threadIdx.x >> 6;                 // k sub-row 0..3 (steps by 4)
    const int scc = (n0 + sc < Nc) ? (n0 + sc) : (Nc - 1);   // clamped col
    const bool scvalid = (n0 + sc < Nc);
    float* shw = &bsh[sc * BSTRIDE];                 // this thread's LDS column

    v8f acc0 = {}, acc1 = {}, acc2 = {}, acc3 = {};

    const int Kmain = K & ~(KT - 1);
    const float* Ap = Arow + half2;

    // ---- Main K tiles: no K guards anywhere -------------------------------
    for (int k0 = 0; k0 < Kmain; k0 += KT) {
        __syncthreads();
#pragma unroll
        for (int k = sk; k < KT; k += 4) {           // 8 staged elems / thread
            float v = Bm[(size_t)(k0 + k) * Nc + scc];
            if (!scvalid) v = 0.0f;                  // cndmask, no exec branch
            shw[k] = v;
        }
        __syncthreads();
#pragma unroll
        for (int kt = 0; kt < KT; kt += 4) {
            const v2f a  = *(const v2f*)(Ap + kt);   // immediate-offset loads
            const int kb = kt + half2;
            const v2f b0 = *(const v2f*)(&bsh[(l15)      * BSTRIDE + kb]);
            const v2f b1 = *(const v2f*)(&bsh[(16 + l15) * BSTRIDE + kb]);
            const v2f b2 = *(const v2f*)(&bsh[(32 + l15) * BSTRIDE + kb]);
            const v2f b3 = *(const v2f*)(&bsh[(48 + l15) * BSTRIDE + kb]);
            acc0 = wmma4(a, b0, acc0);
            acc1 = wmma4(a, b1, acc1);
            acc2 = wmma4(a, b2, acc2);
            acc3 = wmma4(a, b3, acc3);
        }
        Ap += KT;
    }

    // ---- Tail K tile (only when K % KT != 0): clamped, zero-padded --------
    if (Kmain < K) {
        __syncthreads();
#pragma unroll
        for (int k = sk; k < KT; k += 4) {
            const int gk = Kmain + k;
            const int ck = (gk < K) ? gk : (K - 1);
            float v = Bm[(size_t)ck * Nc + scc];
            if (!scvalid || gk >= K) v = 0.0f;
            shw[k] = v;
        }
        __syncthreads();
#pragma unroll
        for (int kt = 0; kt < KT; kt += 4) {
            int ka = Kmain + kt + half2;
            ka = (ka < K) ? ka : 0;                  // B is zero there
            const v2f a  = *(const v2f*)(Arow + ka);
            const int kb = kt + half2;
            const v2f b0 = *(const v2f*)(&bsh[(l15)      * BSTRIDE + kb]);
            const v2f b1 = *(const v2f*)(&bsh[(16 + l15) * BSTRIDE + kb]);
            const v2f b2 = *(const v2f*)(&bsh[(32 + l15) * BSTRIDE + kb]);
            const v2f b3 = *(const v2f*)(&bsh[(48 + l15) * BSTRIDE + kb]);
            acc0 = wmma4(a, b0, acc0);
            acc1 = wmma4(a, b1, acc1);
            acc2 = wmma4(a, b2, acc2);
            acc3 = wmma4(a, b3, acc3);
        }
    }

    if (!validM) return;
    const v8f accs[4] = {acc0, acc1, acc2, acc3};
#pragma unroll
    for (int j = 0; j < 4; ++j) {
        const int col = n0 + j * 16 + l15;
        if (col >= Nc) continue;
        const float bv = BIAS ? bias[col] : 0.0f;
#pragma unroll
        for (int r = 0; r < 8; ++r) {
            const int row = tileM * 16 + (half2 << 2) + r;   // half2<<2 == half*8
            float v = accs[j][r] + bv;
            if (RELU) v = fmaxf(v, 0.0f);
            D[(size_t)row * Nc + col] = v;
        }
    }
}

// ---------------------------------------------------------------------------
// Attention logits: als[n,h] = sum_c h[n,h,c]*a_src[h,c]; same for ald.
// ---------------------------------------------------------------------------
__global__ void gat_attn_logits(const float* __restrict__ h,
                                const float* __restrict__ a_s,
                                const float* __restrict__ a_d,
                                float* __restrict__ als,
                                float* __restrict__ ald,
                                int Heads, int C) {
    const int idx = blockIdx.x * blockDim.x + threadIdx.x;
    if (idx >= NN * Heads) return;
    const int n = idx / Heads, hd = idx % Heads;
    const float* hp = h + (size_t)n * Heads * C + (size_t)hd * C;
    const float* sp = a_s + (size_t)hd * C;
    const float* dp = a_d + (size_t)hd * C;
    float s = 0.f, d = 0.f;
    for (int c = 0; c < C; ++c) {
        const float v = hp[c];
        s += v * sp[c];
        d += v * dp[c];
    }
    als[idx] = s;
    ald[idx] = d;
}

// Monotone float<->uint encoding for atomicMax-based segment max.
static __device__ __forceinline__ unsigned encf(float x) {
    unsigned u = __float_as_uint(x);
    return (u & 0x80000000u) ? ~u : (u | 0x80000000u);
}
static __device__ __forceinline__ float decf(unsigned e) {
    return (e & 0x80000000u) ? __uint_as_float(e & 0x7FFFFFFFu)
                             : __uint_as_float(~e);
}

// e = leaky_relu(als[src] + ald[dst]); segment max into menc[dst].
__global__ void gat_edge_max(const float* __restrict__ als,
                             const float* __restrict__ ald,
                             const int* __restrict__ src,
                             const int* __restrict__ dst,
                             float* __restrict__ ebuf,
                             unsigned* __restrict__ menc,
                             int Heads) {
    const int idx = blockIdx.x * blockDim.x + threadIdx.x;
    if (idx >= EE * Heads) return;
    const int e = idx / Heads, hd = idx % Heads;
    const int s = src[e], d = dst[e];
    float v = als[s * Heads + hd] + ald[d * Heads + hd];
    v = (v > 0.f) ? v : 0.2f * v;           // leaky_relu slope 0.2
    ebuf[idx] = v;
    atomicMax(&menc[d * Heads + hd], encf(v));
}

// ex = exp(e - m[dst]); segment sum into denom[dst].
__global__ void gat_edge_exp(float* __restrict__ ebuf,
                             const unsigned* __restrict__ menc,
                             float* __restrict__ denom,
                             const int* __restrict__ dst,
                             int Heads) {
    const int idx = blockIdx.x * blockDim.x + threadIdx.x;
    if (idx >= EE * Heads) return;
    const int e = idx / Heads, hd = idx % Heads;
    const float m = decf(menc[dst[e] * Heads + hd]);
    const float ex = expf(ebuf[idx] - m);
    ebuf[idx] = ex;
    atomicAdd(&denom[dst[e] * Heads + hd], ex);
}

// out[dst] += h[src] * (ex / denom[dst]) per channel. One block per edge.
__global__ void gat_edge_agg(const float* __restrict__ h,
                             const float* __restrict__ ebuf,
                             const float* __restrict__ denom,
                             const int* __restrict__ src,
                             const int* __restrict__ dst,
                             float* __restrict__ out,
                             int Heads, int C) {
    const int e = blockIdx.x;
    if (e >= EE) return;
    const int s = src[e], d = dst[e];
    const int HCl = Heads * C;
    const float* hp = h + (size_t)s * HCl;
    float* op = out + (size_t)d * HCl;
    for (int j = threadIdx.x; j < HCl; j += blockDim.x) {
        const int hd = j / C;
        const float coef = ebuf[(size_t)e * Heads + hd] /
                           denom[(size_t)d * Heads + hd];
        atomicAdd(&op[j], hp[j] * coef);
    }
}

// buf = (ELU ? elu : id)(buf + bias[col])
template <bool ELU>
__global__ void gat_bias_act(float* __restrict__ buf,
                             const float* __restrict__ bias,
                             int total, int cols) {
    const int i = blockIdx.x * blockDim.x + threadIdx.x;
    if (i >= total) return;
    float v = buf[i] + bias[i % cols];
    if (ELU) v = (v > 0.f) ? v : expm1f(v);
    buf[i] = v;
}

// ---------------------------------------------------------------------------
// Host orchestration
// ---------------------------------------------------------------------------
static inline void launch_gemm(const float* A, const float* B, float* D,
                               int M, int Nc, int K, hipStream_t s) {
    dim3 grid((Nc + 63) / 64, ((M + 15) / 16 + 7) / 8);
    gat_gemm_wmma<false, false><<<grid, dim3(256), 0, s>>>(A, B, nullptr, D, M, Nc, K);
}

extern "C" void kernel_launch(void* const* d_in, const int* in_sizes, int n_in,
                              void* d_out, int out_size, void* d_ws, size_t ws_size,
                              hipStream_t stream) {
    const float* x   = (const float*)d_in[0];
    const int*   ei  = (const int*)d_in[1];   // [2,E] int32 (JAX default x64 off)
    const int*   src = ei;
    const int*   dst = ei + EE;
    const float* W1  = (const float*)d_in[2];
    const float* a1s = (const float*)d_in[3];
    const float* a1d = (const float*)d_in[4];
    const float* b1  = (const float*)d_in[5];
    const float* W2  = (const float*)d_in[6];
    const float* a2s = (const float*)d_in[7];
    const float* a2d = (const float*)d_in[8];
    const float* b2  = (const float*)d_in[9];
    const float* W3  = (const float*)d_in[10];
    const float* a3s = (const float*)d_in[11];
    const float* a3d = (const float*)d_in[12];
    const float* b3  = (const float*)d_in[13];
    const float* fcW = (const float*)d_in[14];
    const float* fcb = (const float*)d_in[15];
    float* out = (float*)d_out;

    // Workspace carve-out (~172 MB)
    char* p = (char*)d_ws;
    auto carve = [&](size_t bytes) -> char* {
        char* r = p;
        p += (bytes + 255) & ~(size_t)255;
        return r;
    };
    float*    hbuf  = (float*)carve((size_t)NN * HC * 4);   // pre-aggregation features
    float*    obuf  = (float*)carve((size_t)NN * HC * 4);   // aggregated output
    float*    als   = (float*)carve((size_t)NN * HH * 4);
    float*    ald   = (float*)carve((size_t)NN * HH * 4);
    unsigned* menc  = (unsigned*)carve((size_t)NN * HH * 4);
    float*    denom = (float*)carve((size_t)NN * HH * 4);
    float*    ebuf  = (float*)carve((size_t)EE * HH * 4);
    float*    h3    = (float*)carve((size_t)NN * CCH * 4);
    float*    d3    = (float*)carve((size_t)NN * CCH * 4);

    const int TPB = 256;
    auto run_attention = [&](const float* hfeat, const float* as_, const float* ad_,
                             float* aggbuf, int Heads, int C) {
        const int nh = NN * Heads, eh = EE * Heads;
        hipMemsetAsync(menc, 0, (size_t)nh * 4, stream);
        hipMemsetAsync(denom, 0, (size_t)nh * 4, stream);
        hipMemsetAsync(aggbuf, 0, (size_t)NN * Heads * C * 4, stream);
        gat_attn_logits<<<(nh + TPB - 1) / TPB, TPB, 0, stream>>>(hfeat, as_, ad_, als, ald, Heads, C);
        gat_edge_max<<<(eh + TPB - 1) / TPB, TPB, 0, stream>>>(als, ald, src, dst, ebuf, menc, Heads);
        gat_edge_exp<<<(eh + TPB - 1) / TPB, TPB, 0, stream>>>(ebuf, menc, denom, dst, Heads);
        gat_edge_agg<<<EE, TPB, 0, stream>>>(hfeat, ebuf, denom, src, dst, aggbuf, Heads, C);
    };

    // ---- Layer 1: h1 = elu(GAT(x; W1) + b1), concat=True -> [N, 1920]
    launch_gemm(x, W1, hbuf, NN, HC, FIN, stream);
    run_attention(hbuf, a1s, a1d, obuf, HH, CCH);
    gat_bias_act<true><<<((NN * HC) + TPB - 1) / TPB, TPB, 0, stream>>>(obuf, b1, NN * HC, HC);

    // ---- Layer 2: h2 = elu(GAT(h1; W2) + b2) -> [N, 1920]
    launch_gemm(obuf, W2, hbuf, NN, HC, HC, stream);
    run_attention(hbuf, a2s, a2d, obuf, HH, CCH);   // obuf re-zeroed inside
    gat_bias_act<true><<<((NN * HC) + TPB - 1) / TPB, TPB, 0, stream>>>(obuf, b2, NN * HC, HC);

    // ---- Layer 3: single head, concat=False (mean over 1 head == identity) -> [N, 120]
    launch_gemm(obuf, W3, h3, NN, CCH, HC, stream);
    run_attention(h3, a3s, a3d, d3, 1, CCH);
    gat_bias_act<false><<<((NN * CCH) + TPB - 1) / TPB, TPB, 0, stream>>>(d3, b3, NN * CCH, CCH);

    // ---- Final FC: out = relu(d3 @ fcW + fcb) -> [N, N], written once (400 MB)
    {
        dim3 grid((NN + 63) / 64, ((NN + 15) / 16 + 7) / 8);
        gat_gemm_wmma<true, true><<<grid, dim3(256), 0, stream>>>(d3, fcW, fcb, out, NN, NN, CCH);
    }
}